// AdaptiveAdjacencyMatrix_61486751809979
// MI455X (gfx1250) — compile-verified
//
#include <hip/hip_runtime.h>
#include <hip/hip_bf16.h>
#include <math.h>

typedef __attribute__((ext_vector_type(2))) float v2f;
typedef __attribute__((ext_vector_type(4))) float v4f;
typedef __attribute__((ext_vector_type(8))) float v8f;

#define BATCH 4
#define N 4096
#define D 256
#define NROWS (BATCH * N)   // 16384 dot products

// ---------------------------------------------------------------------------
// Kernel 1: s[row] = dot(h[row, :], w)  via V_WMMA_F32_16X16X4_F32.
// One wave (32 lanes) computes 16 rows. A = 16x4 tile of h, B = 4x16 tile with
// the w-chunk replicated across all 16 columns, so D[m, n] = partial dot for
// every n. Accumulate over D/4 = 64 K-chunks in the f32 accumulator.
//
// 32-bit A 16x4 layout (ISA 7.12.2): lane L<16 -> M=L, VGPR0=K0, VGPR1=K1;
//                                    lane L>=16 -> M=L-16, VGPR0=K2, VGPR1=K3.
// 32-bit B 4x16 layout mirrors the K striping (lanes 0-15: K0/K1 rows in
// v0/v1, lanes 16-31: K2/K3); with w replicated over columns, each lane loads
// w[k0+koff], w[k0+koff+1] where koff = (lane>>4)*2 — same indexing as A.
//
// C/D 16x16 layout: VGPR r, lanes 0-15 -> (M=r, N=lane); lanes 16-31 ->
// (M=8+r, N=lane-16).  Lane 0 holds D[0..7, 0], lane 16 holds D[8..15, 0].
// ---------------------------------------------------------------------------
__global__ __launch_bounds__(128)
void gemv_wmma_kernel(const float* __restrict__ h,
                      const float* __restrict__ w,
                      float* __restrict__ s_out) {
    const int lane    = threadIdx.x & 31;
    const int wave    = blockIdx.x * (blockDim.x >> 5) + (threadIdx.x >> 5);
    const int rowbase = wave * 16;

    const int half = lane >> 4;     // 0: K pair {0,1}; 1: K pair {2,3}
    const int m    = lane & 15;     // row within the 16-row tile

    const float* __restrict__ hrow = h + (size_t)(rowbase + m) * D;

    v8f acc = {};
    #pragma unroll 4
    for (int k0 = 0; k0 < D; k0 += 4) {
        const int koff = k0 + half * 2;
        v2f a, bb;
        a.x  = hrow[koff];
        a.y  = hrow[koff + 1];
        bb.x = w[koff];
        bb.y = w[koff + 1];
        acc = __builtin_amdgcn_wmma_f32_16x16x4_f32(
            /*neg_a=*/false, a, /*neg_b=*/false, bb,
            /*c_mod=*/(short)0, acc, /*reuse_a=*/false, /*reuse_b=*/false);
    }

    // Lanes 0 and 16 hold column N=0 for rows M=0..7 and M=8..15 respectively.
    if (m == 0) {
        const int base = rowbase + half * 8;
        #pragma unroll
        for (int r = 0; r < 8; ++r) s_out[base + r] = acc[r];
    }
}

// ---------------------------------------------------------------------------
// Kernel 2: per-batch softmax over the 4096 scores. One 1024-thread block per
// batch; classic shared-memory tree reduction (wave32-safe, no lane-mask
// assumptions).
// ---------------------------------------------------------------------------
__global__ __launch_bounds__(1024)
void softmax_kernel(const float* __restrict__ s,
                    float* __restrict__ p) {
    const int b = blockIdx.x;
    const int t = threadIdx.x;                 // 0..1023, 4 elems each
    const float* __restrict__ sb = s + b * N;

    __shared__ float red[1024];

    // --- max reduction ---
    float lm = -INFINITY;
    #pragma unroll
    for (int i = 0; i < 4; ++i) lm = fmaxf(lm, sb[t + i * 1024]);
    red[t] = lm;
    __syncthreads();
    for (int off = 512; off > 0; off >>= 1) {
        if (t < off) red[t] = fmaxf(red[t], red[t + off]);
        __syncthreads();
    }
    const float gmax = red[0];
    __syncthreads();

    // --- sum of exp ---
    float ls = 0.0f;
    float e[4];
    #pragma unroll
    for (int i = 0; i < 4; ++i) {
        e[i] = __expf(sb[t + i * 1024] - gmax);
        ls += e[i];
    }
    red[t] = ls;
    __syncthreads();
    for (int off = 512; off > 0; off >>= 1) {
        if (t < off) red[t] += red[t + off];
        __syncthreads();
    }
    const float inv = 1.0f / red[0];

    #pragma unroll
    for (int i = 0; i < 4; ++i) p[b * N + t + i * 1024] = e[i] * inv;
}

// ---------------------------------------------------------------------------
// Kernel 3: broadcast write — the bandwidth-bound part (268 MB of stores).
// Row (b,i) of the output is 4096 copies of p[b*N+i]. One 256-thread block
// per row; each thread stores 4 x float4 (global_store_b128), nontemporal
// since the stream (268 MB) exceeds L2 (192 MB).
// ---------------------------------------------------------------------------
__global__ __launch_bounds__(256)
void broadcast_kernel(const float* __restrict__ p,
                      v4f* __restrict__ out) {
    const int row = blockIdx.x;                // 0..16383  (= b*N + i)
    const float v = p[row];                    // uniform per block -> s_load
    const v4f f = {v, v, v, v};

    v4f* __restrict__ o = out + (size_t)row * (N / 4);
    const int t = threadIdx.x;                 // 0..255
    __builtin_nontemporal_store(f, &o[t]);
    __builtin_nontemporal_store(f, &o[t + 256]);
    __builtin_nontemporal_store(f, &o[t + 512]);
    __builtin_nontemporal_store(f, &o[t + 768]);
}

extern "C" void kernel_launch(void* const* d_in, const int* in_sizes, int n_in,
                              void* d_out, int out_size, void* d_ws, size_t ws_size,
                              hipStream_t stream) {
    const float* h = (const float*)d_in[0];   // [4, 4096, 256]
    const float* w = (const float*)d_in[1];   // [256]
    // d_in[2] is the bias: it is constant along the softmax axis and cancels.

    float* s = (float*)d_ws;                  // [16384]
    float* p = s + NROWS;                     // [16384]
    float* out = (float*)d_out;               // [4, 4096, 4096]

    // 16384 rows / 16 rows-per-wave = 1024 waves; 4 waves per 128-thread block.
    gemv_wmma_kernel<<<NROWS / 16 / 4, 128, 0, stream>>>(h, w, s);

    softmax_kernel<<<BATCH, 1024, 0, stream>>>(s, p);

    broadcast_kernel<<<NROWS, 256, 0, stream>>>(p, (v4f*)out);
}